// GConvGRUClassifier_73083163508914
// MI455X (gfx1250) — compile-verified
//
#include <hip/hip_runtime.h>

#define N_NODES 50000
#define N_EDGES 800000
#define FDIM 128
#define HDIM 128
#define TDIM 10
#define NRB (N_NODES / 16)   // 3125 row-blocks of 16, exact

typedef __attribute__((ext_vector_type(2))) float v2f;
typedef __attribute__((ext_vector_type(8))) float v8f;

// ---------------------------------------------------------------------------
// Scratch zeroing (covers deg + cnt; 32-bit zeros, stored as float4)
// ---------------------------------------------------------------------------
__global__ void zero_f4(float4* p, long n4) {
    long i = blockIdx.x * (long)blockDim.x + threadIdx.x;
    long stride = (long)gridDim.x * blockDim.x;
    float4 z = make_float4(0.f, 0.f, 0.f, 0.f);
    for (; i < n4; i += stride) p[i] = z;
}

// ---------------------------------------------------------------------------
// deg[src[e]] += w[e]   (segment_sum over src)
// ---------------------------------------------------------------------------
__global__ void deg_kernel(const long long* __restrict__ ei,
                           const float* __restrict__ w,
                           float* __restrict__ deg) {
    int e = blockIdx.x * blockDim.x + threadIdx.x;
    if (e >= N_EDGES) return;
    atomicAdd(&deg[(int)ei[e]], w[e]);
}

// ---------------------------------------------------------------------------
// norm[e] = -dinv[src] * w[e] * dinv[dst],  dinv = deg>0 ? 1/sqrt(deg) : 0
// ---------------------------------------------------------------------------
__global__ void norm_kernel(const long long* __restrict__ ei,
                            const float* __restrict__ w,
                            const float* __restrict__ deg,
                            float* __restrict__ nrm) {
    int e = blockIdx.x * blockDim.x + threadIdx.x;
    if (e >= N_EDGES) return;
    int src = (int)ei[e];
    int dst = (int)ei[N_EDGES + e];
    float ds = deg[src], dd = deg[dst];
    float is = ds > 0.f ? 1.f / sqrtf(ds) : 0.f;
    float id = dd > 0.f ? 1.f / sqrtf(dd) : 0.f;
    nrm[e] = -is * w[e] * id;
}

// ---------------------------------------------------------------------------
// CSR build: histogram of dst, exclusive scan, slot scatter
// ---------------------------------------------------------------------------
__global__ void hist_kernel(const long long* __restrict__ ei, int* __restrict__ cnt) {
    int e = blockIdx.x * blockDim.x + threadIdx.x;
    if (e >= N_EDGES) return;
    atomicAdd(&cnt[(int)ei[N_EDGES + e]], 1);
}

__global__ __launch_bounds__(1024) void exscan_kernel(
    const int* __restrict__ cnt, int* __restrict__ off, int* __restrict__ pos) {
    __shared__ int buf[1024];
    __shared__ int carry;
    int tid = threadIdx.x;
    if (tid == 0) carry = 0;
    __syncthreads();
    for (int base = 0; base < N_NODES; base += 1024) {
        int idx = base + tid;
        int v = (idx < N_NODES) ? cnt[idx] : 0;
        buf[tid] = v;
        __syncthreads();
        for (int ofs = 1; ofs < 1024; ofs <<= 1) {
            int t = (tid >= ofs) ? buf[tid - ofs] : 0;
            __syncthreads();
            buf[tid] += t;
            __syncthreads();
        }
        int incl = buf[tid];
        int excl = incl - v + carry;           // reads carry
        if (idx < N_NODES) { off[idx] = excl; pos[idx] = excl; }
        __syncthreads();
        if (tid == 1023) carry += incl;        // chunk total
        __syncthreads();
    }
    if (tid == 0) off[N_NODES] = carry;        // == N_EDGES
}

__global__ void scatter_kernel(const long long* __restrict__ ei,
                               const float* __restrict__ nrm,
                               int* __restrict__ pos,
                               int* __restrict__ csr_src,
                               float* __restrict__ csr_nrm) {
    int e = blockIdx.x * blockDim.x + threadIdx.x;
    if (e >= N_EDGES) return;
    int dst = (int)ei[N_EDGES + e];
    int slot = atomicAdd(&pos[dst], 1);
    csr_src[slot] = (int)ei[e];
    csr_nrm[slot] = nrm[e];
}

// ---------------------------------------------------------------------------
// Atomic-free propagation: one wave32 per destination node.
// Each lane owns 4 features; gathers norm*tin[src] rows, one streamed store.
// Nodes with no in-edges correctly write zeros (segment_sum semantics).
// ---------------------------------------------------------------------------
__global__ void prop_csr_kernel(const int* __restrict__ off,
                                const int* __restrict__ csr_src,
                                const float* __restrict__ csr_nrm,
                                const float* __restrict__ tin,
                                float* __restrict__ tout) {
    int wave = (blockIdx.x * blockDim.x + threadIdx.x) >> 5;
    int lane = threadIdx.x & 31;
    if (wave >= N_NODES) return;
    int beg = off[wave], end = off[wave + 1];
    float4 acc = make_float4(0.f, 0.f, 0.f, 0.f);
    for (int i = beg; i < end; ++i) {
        int src = csr_src[i];       // wave-uniform -> broadcast load
        float s = csr_nrm[i];
        float4 v = ((const float4*)(tin + (long)src * FDIM))[lane];
        acc.x += s * v.x;
        acc.y += s * v.y;
        acc.z += s * v.z;
        acc.w += s * v.w;
    }
    ((float4*)(tout + (long)wave * FDIM))[lane] = acc;
}

// ---------------------------------------------------------------------------
// Weight pre-swizzle: interleave row pairs so a B fragment is one b64 load.
// Wp[gate][ksel][pair][col][par] = W_gate[ksel][2*pair+par][col]
// ---------------------------------------------------------------------------
__global__ void wprep_kernel(const float* __restrict__ Wxz,
                             const float* __restrict__ Wxh,
                             float* __restrict__ Wp) {
    int i = blockIdx.x * blockDim.x + threadIdx.x;   // 2*3*128*128 threads
    if (i >= 2 * 3 * FDIM * HDIM) return;
    int col  = i & (HDIM - 1);
    int row  = (i >> 7) & (FDIM - 1);
    int ksel = (i >> 14) % 3;
    int gate = i / (3 * FDIM * HDIM);
    const float* W = gate ? Wxh : Wxz;
    float v = W[(long)ksel * FDIM * HDIM + (long)row * HDIM + col];
    int pair = row >> 1, par = row & 1;
    Wp[(long)(gate * 3 + ksel) * 16384 + (long)pair * (HDIM * 2) + col * 2 + par] = v;
}

// ---------------------------------------------------------------------------
// K=128 GEMM slab with V_WMMA_F32_16X16X4_F32.
// A frag (16x4 f32): lane m(0-15): {A[m][k], A[m][k+1]}, lane 16+m: {A[m][k+2], A[m][k+3]}
// B frag (4x16):     lane n(0-15): {B[k][n], B[k+1][n]}, lane 16+n: {B[k+2][n], B[k+3][n]}
// B from the pre-swizzled Wp slab -> one contiguous b64 per lane.
// MODE 0: A = a0 rows;  MODE 2: A = 2*a1 - a0 (Chebyshev Tx2 on the fly)
// ---------------------------------------------------------------------------
template <int MODE>
__device__ __attribute__((always_inline)) void gemm_k128(
    v8f (&acc)[8],
    const float* __restrict__ a0, const float* __restrict__ a1,
    const float* __restrict__ Wp, int mlane, int khalf) {
#pragma unroll 4
    for (int kk = 0; kk < 32; ++kk) {
        int acol = kk * 4 + khalf * 2;               // even -> 8B aligned
        v2f a;
        if (MODE == 2) {
            v2f ax = *(const v2f*)(a0 + acol);
            v2f ap = *(const v2f*)(a1 + acol);
            a.x = 2.f * ap.x - ax.x;
            a.y = 2.f * ap.y - ax.y;
        } else {
            a = *(const v2f*)(a0 + acol);
        }
        int pair = kk * 2 + khalf;                   // row pair index
        const float* Wk = Wp + (long)pair * (HDIM * 2);
#pragma unroll
        for (int nt = 0; nt < 8; ++nt) {
            int col = nt * 16 + mlane;
            v2f b = *(const v2f*)(Wk + col * 2);     // {W[2p][col], W[2p+1][col]}
            acc[nt] = __builtin_amdgcn_wmma_f32_16x16x4_f32(
                false, a, false, b, (short)0, acc[nt], false, false);
        }
    }
}

// ---------------------------------------------------------------------------
// Fused: preact = Tx0@W[0] + Tx1@W[1] + Tx2@W[2] + bias for gates z,h;
// Hn = (1 - sigmoid(z)) * tanh(h);  out = Hn @ Wlin + blin.
// One wave per 16-row block; Hn tile staged in LDS (intra-wave only).
// ---------------------------------------------------------------------------
__global__ __launch_bounds__(256) void gru_gemm_kernel(
    const float* __restrict__ x, const float* __restrict__ tx1,
    const float* __restrict__ p2, const float* __restrict__ Wp,
    const float* __restrict__ bxz, const float* __restrict__ bhz,
    const float* __restrict__ bxh, const float* __restrict__ bhh,
    const float* __restrict__ Wlin, const float* __restrict__ blin,
    float* __restrict__ out) {
    __shared__ float lds[8][16][HDIM];  // 8 waves x 8KB = 64KB
    int wib  = threadIdx.x >> 5;
    int lane = threadIdx.x & 31;
    int rb = blockIdx.x * 8 + wib;
    if (rb >= NRB) return;  // uniform per wave -> EXEC all-ones for WMMA
    int row0 = rb * 16;

    float (*Zt)[HDIM] = lds[wib];
    int mlane = lane & 15;
    int khalf = lane >> 4;

    const float* arow_x  = x   + (long)(row0 + mlane) * FDIM;
    const float* arow_t1 = tx1 + (long)(row0 + mlane) * FDIM;
    const float* arow_p2 = p2  + (long)(row0 + mlane) * FDIM;

    for (int gate = 0; gate < 2; ++gate) {
        const float* Wg = Wp + (long)(gate * 3) * 16384;
        const float* b0 = gate ? bxh : bxz;
        const float* b1 = gate ? bhh : bhz;

        v8f acc[8];
#pragma unroll
        for (int i = 0; i < 8; ++i) acc[i] = (v8f){0.f, 0.f, 0.f, 0.f, 0.f, 0.f, 0.f, 0.f};

        gemm_k128<0>(acc, arow_x,  nullptr, Wg,             mlane, khalf); // Tx0
        gemm_k128<0>(acc, arow_t1, nullptr, Wg + 16384,     mlane, khalf); // Tx1
        gemm_k128<2>(acc, arow_x,  arow_p2, Wg + 2 * 16384, mlane, khalf); // Tx2 = 2*P2 - x

#pragma unroll
        for (int nt = 0; nt < 8; ++nt) {
            int col = nt * 16 + mlane;
            float bias = b0[col] + b1[col];
#pragma unroll
            for (int r = 0; r < 8; ++r) {
                int m = khalf * 8 + r;  // C/D layout: lanes 16-31 hold M=8..15
                float v = acc[nt][r] + bias;
                if (gate == 0) {
                    Zt[m][col] = 1.f / (1.f + __expf(-v));          // Z
                } else {
                    Zt[m][col] = (1.f - Zt[m][col]) * tanhf(v);     // Hn
                }
            }
        }
    }

    // Projection 128 -> 10 : 160 outputs per row-block, 5 per lane.
#pragma unroll
    for (int j = 0; j < 5; ++j) {
        int o = lane * 5 + j;  // 0..159
        int m = o / TDIM, t = o % TDIM;
        float acc = blin[t];
#pragma unroll 8
        for (int n = 0; n < HDIM; ++n) acc += Zt[m][n] * Wlin[n * TDIM + t];
        out[(long)(row0 + m) * TDIM + t] = acc;
    }
}

// ---------------------------------------------------------------------------
extern "C" void kernel_launch(void* const* d_in, const int* in_sizes, int n_in,
                              void* d_out, int out_size, void* d_ws, size_t ws_size,
                              hipStream_t stream) {
    const float*     x    = (const float*)d_in[0];
    const long long* ei   = (const long long*)d_in[1];
    const float*     w    = (const float*)d_in[2];
    const float*     Wxz  = (const float*)d_in[3];
    const float*     bxz  = (const float*)d_in[4];
    const float*     bhz  = (const float*)d_in[6];
    const float*     Wxh  = (const float*)d_in[11];
    const float*     bxh  = (const float*)d_in[12];
    const float*     bhh  = (const float*)d_in[14];
    const float*     Wlin = (const float*)d_in[15];
    const float*     blin = (const float*)d_in[16];
    float* out = (float*)d_out;

    // ws layout (32-bit words; float4-aligned section starts)
    float* ws      = (float*)d_ws;
    float* deg     = ws;                                   // [0,       50000)
    int*   cnt     = (int*)(ws + 50000);                   // [50000,  100000)
    int*   off     = (int*)(ws + 100000);                  // N+1 ints (padded)
    int*   pos     = (int*)(ws + 150016);                  // N ints (padded)
    float* nrm     = ws + 200016;                          // E floats
    int*   csr_src = (int*)(ws + 1000016);                 // E ints
    float* csr_nrm = ws + 1800016;                         // E floats
    float* tx1     = ws + 2600016;                         // N*128 (16B aligned)
    float* p2      = tx1 + (size_t)N_NODES * FDIM;         // N*128
    float* Wp      = p2  + (size_t)N_NODES * FDIM;         // 98304

    // zero deg + cnt (contiguous 100000 words)
    zero_f4<<<256, 256, 0, stream>>>((float4*)deg, 100000 / 4);

    wprep_kernel<<<(2 * 3 * FDIM * HDIM + 255) / 256, 256, 0, stream>>>(Wxz, Wxh, Wp);

    deg_kernel <<<(N_EDGES + 255) / 256, 256, 0, stream>>>(ei, w, deg);
    norm_kernel<<<(N_EDGES + 255) / 256, 256, 0, stream>>>(ei, w, deg, nrm);

    hist_kernel   <<<(N_EDGES + 255) / 256, 256, 0, stream>>>(ei, cnt);
    exscan_kernel <<<1, 1024, 0, stream>>>(cnt, off, pos);
    scatter_kernel<<<(N_EDGES + 255) / 256, 256, 0, stream>>>(ei, nrm, pos, csr_src, csr_nrm);

    prop_csr_kernel<<<(N_NODES + 7) / 8, 256, 0, stream>>>(off, csr_src, csr_nrm, x,   tx1);
    prop_csr_kernel<<<(N_NODES + 7) / 8, 256, 0, stream>>>(off, csr_src, csr_nrm, tx1, p2);

    gru_gemm_kernel<<<(NRB + 7) / 8, 256, 0, stream>>>(
        x, tx1, p2, Wp, bxz, bhz, bxh, bhh, Wlin, blin, out);
}